// GCN_65206193488468
// MI455X (gfx1250) — compile-verified
//
#include <hip/hip_runtime.h>
#include <hip/hip_bf16.h>
#include <math.h>

// ---------------------------------------------------------------------------
// GCN forward for MI455X (gfx1250, wave32).
//   N=50000 nodes, E=600000 edges, D=128, C=10, L=3
// GEMMs: V_WMMA_F32_16X16X4_F32, K=128 fully unrolled (32 WMMA/wave).
//   B staged TRANSPOSED in LDS (padded stride 132) -> 1x ds_load_b64 per WMMA,
//   bank-conflict-free.
// Aggregation (memory-bound, ~2.7GB + ~256M atomics): wave-per-edge float4
//   gathers + native global_atomic_add_f32 via unsafeAtomicAdd.
// ---------------------------------------------------------------------------

#define GCN_N 50000
#define GCN_E 600000
#define GCN_D 128
#define GCN_C 10
#define GCN_L 3
#define GCN_EPS 1e-5f
#define KP 132   // padded LDS row stride (floats) for transposed W

typedef __attribute__((ext_vector_type(2))) float v2f;
typedef __attribute__((ext_vector_type(8))) float v8f;

__device__ __forceinline__ void atomAddF32(float* p, float v) {
  unsafeAtomicAdd(p, v);   // lowers to global_atomic_add_f32 (no CAS loop)
}

// ---------------------------------------------------------------- utilities
__global__ void k_fill4(float4* __restrict__ p, float v, int n4) {
  int i = blockIdx.x * blockDim.x + threadIdx.x;
  if (i < n4) p[i] = make_float4(v, v, v, v);
}

__global__ void k_deg_accum(float* __restrict__ deg, const int* __restrict__ dst, int e) {
  int i = blockIdx.x * blockDim.x + threadIdx.x;
  if (i < e) atomAddF32(&deg[dst[i]], 1.0f);
}

__global__ void k_rsqrt_inplace(float* __restrict__ p, int n) {
  int i = blockIdx.x * blockDim.x + threadIdx.x;
  if (i < n) p[i] = rsqrtf(p[i]);   // deg >= 1 always (self loop)
}

// ------------------------------------------------------------- WMMA GEMM f32
// C[rows x Dout] = act(A[rows x 128] @ Wpad[128 x Dout] + bias)
// W has Wcols real columns; zero-padded to Dout. Staged transposed in LDS:
//   sWt[c * KP + k] = W[k][c]   (KP=132 -> lanes hit distinct bank pairs)
// Block: 32 * (Dout/16) threads, one wave per 16-col tile; grid.x = rows/16.
__global__ void __launch_bounds__(256)
k_gemm_wmma(const float* __restrict__ A, const float* __restrict__ W,
            const float* __restrict__ bias, float* __restrict__ C,
            int Dout, int Wcols, int relu) {
  extern __shared__ float sWt[];                // Dout x KP
  const int K = 128;
  for (int idx = threadIdx.x; idx < K * Dout; idx += blockDim.x) {
    int r = idx / Dout, c = idx - r * Dout;     // source-major: coalesced global
    sWt[c * KP + r] = (c < Wcols) ? W[r * Wcols + c] : 0.0f;
  }
  __syncthreads();

  const int lane = threadIdx.x & 31;
  const int wv   = threadIdx.x >> 5;
  const int m    = lane & 15;        // A row within tile / B,C column
  const int kh   = lane >> 4;        // K-half select (0/1)
  const int colBase = wv * 16;
  const float* arow = A + (size_t)(blockIdx.x * 16 + m) * K;
  // this lane's B column, as float2 pairs (ka, ka+1) with ka even
  const float2* bcol = (const float2*)(sWt + (colBase + m) * KP) + kh;

  v8f acc = {};
#pragma unroll
  for (int kk = 0; kk < K; kk += 4) {
    const int ka = kk + kh * 2;
    v2f av, bv;
    av[0] = arow[ka];
    av[1] = arow[ka + 1];
    float2 b2 = bcol[kk >> 1];       // ds_load_b64: {W[ka][col], W[ka+1][col]}
    bv[0] = b2.x;
    bv[1] = b2.y;
    acc = __builtin_amdgcn_wmma_f32_16x16x4_f32(
        /*neg_a=*/false, av, /*neg_b=*/false, bv,
        /*c_mod=*/(short)0, acc, /*reuse_a=*/false, /*reuse_b=*/false);
  }

  const int rbase = blockIdx.x * 16 + kh * 8;   // D rows: vgpr r -> M=r + 8*kh
#pragma unroll
  for (int r = 0; r < 8; ++r) {
    float v = acc[r];
    if (bias) v += bias[colBase + m];
    if (relu) v = fmaxf(v, 0.0f);
    C[(size_t)(rbase + r) * Dout + colBase + m] = v;
  }
}

// ------------------------------------------------------- edge scatter (D=128)
// One wave per edge (incl. self loops e>=E): agg[dst] += t[src] * dinv_s*dinv_d
__global__ void __launch_bounds__(256)
k_scatter128(const float* __restrict__ t, const int* __restrict__ src,
             const int* __restrict__ dst, const float* __restrict__ dinv,
             float* __restrict__ agg, int E, int total) {
  int gid  = blockIdx.x * blockDim.x + threadIdx.x;
  int e    = gid >> 5;
  int lane = gid & 31;
  if (e >= total) return;
  int s, d;
  if (e < E) { s = src[e]; d = dst[e]; } else { s = d = e - E; }
  float nrm = dinv[s] * dinv[d];
  const float4* tr = (const float4*)(t + (size_t)s * GCN_D);
  float4 v = tr[lane];
  float* ar = agg + (size_t)d * GCN_D + lane * 4;
  atomAddF32(ar + 0, v.x * nrm);
  atomAddF32(ar + 1, v.y * nrm);
  atomAddF32(ar + 2, v.z * nrm);
  atomAddF32(ar + 3, v.w * nrm);
}

// ------------------------------------------- edge scatter (final, 10 of 16 cols)
__global__ void __launch_bounds__(256)
k_scatter10(const float* __restrict__ t, const int* __restrict__ src,
            const int* __restrict__ dst, const float* __restrict__ dinv,
            float* __restrict__ agg, int E, int total) {
  int gid = blockIdx.x * blockDim.x + threadIdx.x;
  int e = gid >> 4;
  int c = gid & 15;
  if (e >= total || c >= GCN_C) return;
  int s, d;
  if (e < E) { s = src[e]; d = dst[e]; } else { s = d = e - E; }
  float nrm = dinv[s] * dinv[d];
  atomAddF32(&agg[(size_t)d * 16 + c], t[(size_t)s * 16 + c] * nrm);
}

// ---------------------------------------------------- batchnorm column stats
// stats[0..127]=sum, stats[128..255]=sumsq. Block: 256 thr = 128 cols x 2 rows.
__global__ void __launch_bounds__(256)
k_colstats(const float* __restrict__ h, float* __restrict__ stats, int nrows) {
  const int ROWS = 128;
  int c  = threadIdx.x & 127;
  int rh = threadIdx.x >> 7;
  float s = 0.0f, q = 0.0f;
  int r0 = blockIdx.x * ROWS + rh;
#pragma unroll 4
  for (int i = 0; i < ROWS; i += 2) {
    int r = r0 + i;
    if (r < nrows) {
      float v = h[(size_t)r * GCN_D + c];
      s += v;
      q += v * v;
    }
  }
  atomAddF32(&stats[c], s);
  atomAddF32(&stats[GCN_D + c], q);
}

// ------------------------------------------------------- batchnorm + relu
__global__ void __launch_bounds__(256)
k_bnrelu(float* __restrict__ h, const float* __restrict__ stats,
         const float* __restrict__ gamma, const float* __restrict__ beta, int n) {
  int i = blockIdx.x * blockDim.x + threadIdx.x;
  if (i >= n) return;
  int c = i & (GCN_D - 1);
  const float invN = 1.0f / (float)GCN_N;
  float mu  = stats[c] * invN;
  float var = stats[GCN_D + c] * invN - mu * mu;
  float rs  = rsqrtf(var + GCN_EPS);
  float v = (h[i] - mu) * rs * gamma[c] + beta[c];
  h[i] = fmaxf(v, 0.0f);
}

// ---------------------------------------------------------- final log-softmax
__global__ void __launch_bounds__(256)
k_logsoftmax(const float* __restrict__ agg /* N x 16 */,
             const float* __restrict__ fb, float* __restrict__ out, int nrows) {
  int r = blockIdx.x * blockDim.x + threadIdx.x;
  if (r >= nrows) return;
  float v[GCN_C];
  float m = -INFINITY;
#pragma unroll
  for (int c = 0; c < GCN_C; ++c) {
    v[c] = agg[(size_t)r * 16 + c] + fb[c];
    m = fmaxf(m, v[c]);
  }
  float sum = 0.0f;
#pragma unroll
  for (int c = 0; c < GCN_C; ++c) sum += __expf(v[c] - m);
  float lse = __logf(sum);
#pragma unroll
  for (int c = 0; c < GCN_C; ++c) out[(size_t)r * GCN_C + c] = v[c] - m - lse;
}

// ---------------------------------------------------------------------------
extern "C" void kernel_launch(void* const* d_in, const int* in_sizes, int n_in,
                              void* d_out, int out_size, void* d_ws, size_t ws_size,
                              hipStream_t stream) {
  const float* x     = (const float*)d_in[0];
  const int*   ei    = (const int*)d_in[1];
  const float* fc1W  = (const float*)d_in[2];
  const float* fc1b  = (const float*)d_in[3];
  const float* convW = (const float*)d_in[4];   // [L,128,128]
  /* conv_b (d_in[5]) cancels exactly inside batchnorm -> unused */
  const float* gamma = (const float*)d_in[6];   // [L,128]
  const float* beta  = (const float*)d_in[7];
  const float* finW  = (const float*)d_in[8];   // [128,10]
  const float* finb  = (const float*)d_in[9];
  float* out = (float*)d_out;

  const int N = GCN_N, E = GCN_E, D = GCN_D;
  const int TOT = E + N;                        // edges + self loops
  const int* src = ei;
  const int* dst = ei + E;

  // ---- workspace carving (needs ~51.5 MB) ----
  char* ws = (char*)d_ws;
  float* dinv  = (float*)(ws);                          // N floats
  float* hA    = (float*)(ws + 200192);                 // N x 128
  float* hB    = (float*)(ws + 200192 + 25600000);      // N x 128 (scratch t)
  float* stats = (float*)(ws + 200192 + 2 * 25600000);  // 256 floats
  (void)ws_size; (void)in_sizes; (void)n_in; (void)out_size;

  const int TB = 256;
  // ---- degree / normalization ----
  k_fill4<<<(N / 4 + TB - 1) / TB, TB, 0, stream>>>((float4*)dinv, 1.0f, N / 4);
  k_deg_accum<<<(E + TB - 1) / TB, TB, 0, stream>>>(dinv, dst, E);
  k_rsqrt_inplace<<<(N + TB - 1) / TB, TB, 0, stream>>>(dinv, N);

  const int rowTiles = N / 16;                           // 3125 exactly
  const size_t shmainsz = (size_t)D * KP * sizeof(float);  // 128*132*4 = 66 KB
  const size_t shfinsz  = (size_t)16 * KP * sizeof(float); // 16*132*4 ~ 8.25 KB

  // ---- fc1: h = relu(x @ fc1_W + fc1_b) -> hA ----
  k_gemm_wmma<<<rowTiles, 256, shmainsz, stream>>>(x, fc1W, fc1b, hA, D, D, 1);

  // ---- 3 GCN layers ----
  for (int l = 0; l < GCN_L; ++l) {
    // t = h @ W_l  (no bias: conv bias cancels in BN)
    k_gemm_wmma<<<rowTiles, 256, shmainsz, stream>>>(hA, convW + (size_t)l * D * D,
                                                     nullptr, hB, D, D, 0);
    // agg = scatter(t) -> hA
    k_fill4<<<(N * D / 4 + TB - 1) / TB, TB, 0, stream>>>((float4*)hA, 0.0f, N * D / 4);
    long long sthreads = (long long)TOT * 32;
    k_scatter128<<<(int)((sthreads + TB - 1) / TB), TB, 0, stream>>>(
        hB, src, dst, dinv, hA, E, TOT);
    // batchnorm + relu (in place on hA)
    k_fill4<<<1, TB, 0, stream>>>((float4*)stats, 0.0f, (2 * D) / 4);
    k_colstats<<<(N + 127) / 128, 256, 0, stream>>>(hA, stats, N);
    k_bnrelu<<<(N * D + TB - 1) / TB, TB, 0, stream>>>(hA, stats,
                                                       gamma + l * D, beta + l * D, N * D);
  }

  // ---- final conv: t10 = h @ final_W (padded to 16 cols) -> hB ----
  k_gemm_wmma<<<rowTiles, 32, shfinsz, stream>>>(hA, finW, nullptr, hB, 16, GCN_C, 0);
  // agg10 -> hA (first N*16 floats)
  k_fill4<<<(N * 16 / 4 + TB - 1) / TB, TB, 0, stream>>>((float4*)hA, 0.0f, N * 16 / 4);
  long long s10 = (long long)TOT * 16;
  k_scatter10<<<(int)((s10 + TB - 1) / TB), TB, 0, stream>>>(hB, src, dst, dinv, hA, E, TOT);

  // ---- + final_b, log_softmax -> d_out ----
  k_logsoftmax<<<(N + TB - 1) / TB, TB, 0, stream>>>(hA, finb, out, N);
}